// RegLoss_69054484185800
// MI455X (gfx1250) — compile-verified
//
#include <hip/hip_runtime.h>
#include <hip/hip_bf16.h>

typedef float v2f __attribute__((ext_vector_type(2)));
typedef float v8f __attribute__((ext_vector_type(8)));

#define N_ROWS 8192
#define DIMS   64
#define TILE   128
#define NBLK   (N_ROWS / TILE)          // 64
#define NTRI   (NBLK * (NBLK + 1) / 2)  // 2080
#define LSTR   68                       // LDS row stride (floats): 4-aligned, conflict-free

// ---------------------------------------------------------------- col means
__global__ void mean_kernel(const float* __restrict__ Xn, const float* __restrict__ Xo,
                            float* __restrict__ means) {
    int col = blockIdx.x & 63;
    int m   = blockIdx.x >> 6;
    const float* X = m ? Xo : Xn;
    __shared__ float red[256];
    float s = 0.f;
    for (int i = threadIdx.x; i < N_ROWS; i += 256) s += X[(size_t)i * DIMS + col];
    red[threadIdx.x] = s;
    __syncthreads();
    for (int off = 128; off > 0; off >>= 1) {
        if (threadIdx.x < off) red[threadIdx.x] += red[threadIdx.x + off];
        __syncthreads();
    }
    if (threadIdx.x == 0) means[m * 64 + col] = red[0] * (1.0f / (float)N_ROWS);
}

// ---------------------------------------------------------------- covariance
__global__ void cov_kernel(const float* __restrict__ Xn, const float* __restrict__ Xo,
                           const float* __restrict__ means, float* __restrict__ cov) {
    int m  = blockIdx.x >> 12;
    int pq = blockIdx.x & 4095;
    int p = pq >> 6, q = pq & 63;
    const float* X = m ? Xo : Xn;
    __shared__ float red[256];
    float s = 0.f;
    for (int i = threadIdx.x; i < N_ROWS; i += 256)
        s += X[(size_t)i * DIMS + p] * X[(size_t)i * DIMS + q];
    red[threadIdx.x] = s;
    __syncthreads();
    for (int off = 128; off > 0; off >>= 1) {
        if (threadIdx.x < off) red[threadIdx.x] += red[threadIdx.x + off];
        __syncthreads();
    }
    if (threadIdx.x == 0)
        cov[m * 4096 + pq] = red[0] * (1.0f / (float)N_ROWS)
                             - means[m * 64 + p] * means[m * 64 + q];
}

// -------------------------------------------- parallel-order Jacobi eigensolver
// One block (64 threads) per 64x64 symmetric matrix. Round-robin pairing:
// 32 disjoint rotations per step, 63 steps per sweep.
__global__ void jacobi_kernel(const float* __restrict__ cov, float* __restrict__ Vout,
                              float* __restrict__ sOut) {
    int m = blockIdx.x;
    const float* C = cov + m * 4096;
    __shared__ float A[64 * 64];
    __shared__ float Vm[64 * 64];
    __shared__ int   perm[64];
    __shared__ float cs[32], sn[32];
    int t = threadIdx.x; // 0..63
    for (int i = t; i < 4096; i += 64) { A[i] = C[i]; Vm[i] = 0.f; }
    __syncthreads();
    Vm[t * 64 + t] = 1.f;
    perm[t] = t;
    __syncthreads();

    for (int sweep = 0; sweep < 10; ++sweep) {
        for (int step = 0; step < 63; ++step) {
            if (t < 32) {
                int p = perm[t], q = perm[63 - t];
                float apq = A[p * 64 + q];
                float c = 1.f, s = 0.f;
                if (fabsf(apq) > 1e-12f) {
                    float tau = (A[q * 64 + q] - A[p * 64 + p]) / (2.f * apq);
                    float tt  = (tau >= 0.f ? 1.f : -1.f) /
                                (fabsf(tau) + sqrtf(1.f + tau * tau));
                    c = rsqrtf(1.f + tt * tt);
                    s = tt * c;
                }
                cs[t] = c; sn[t] = s;
            }
            __syncthreads();
            // rows:  A <- J^T A  (thread t owns column t)
            for (int k = 0; k < 32; ++k) {
                int p = perm[k], q = perm[63 - k];
                float c = cs[k], s = sn[k];
                float ap = A[p * 64 + t], aq = A[q * 64 + t];
                A[p * 64 + t] = c * ap - s * aq;
                A[q * 64 + t] = s * ap + c * aq;
            }
            __syncthreads();
            // cols:  A <- A J ;  V <- V J  (thread t owns row t)
            for (int k = 0; k < 32; ++k) {
                int p = perm[k], q = perm[63 - k];
                float c = cs[k], s = sn[k];
                float ap = A[t * 64 + p], aq = A[t * 64 + q];
                A[t * 64 + p] = c * ap - s * aq;
                A[t * 64 + q] = s * ap + c * aq;
                float vp = Vm[t * 64 + p], vq = Vm[t * 64 + q];
                Vm[t * 64 + p] = c * vp - s * vq;
                Vm[t * 64 + q] = s * vp + c * vq;
            }
            __syncthreads();
            if (t == 0) {
                int last = perm[63];
                for (int i = 63; i > 1; --i) perm[i] = perm[i - 1];
                perm[1] = last;
            }
            __syncthreads();
        }
    }
    for (int i = t; i < 4096; i += 64) Vout[m * 4096 + i] = Vm[i];
    sOut[m * 64 + t] = A[t * 64 + t];
}

// ---------------------------------------------------------------- whiten + sq
// 4 rows x 64 cols per 256-thread block.  W = (X - mu) @ V * rsqrt(s + eps)
__global__ void whiten_kernel(const float* __restrict__ Xn, const float* __restrict__ Xo,
                              const float* __restrict__ means, const float* __restrict__ Vmat,
                              const float* __restrict__ sval,
                              float* __restrict__ Wn, float* __restrict__ Wo,
                              float* __restrict__ sqn, float* __restrict__ sqo) {
    int m  = blockIdx.x >> 11;
    int rb = blockIdx.x & 2047;
    const float* X = m ? Xo : Xn;
    float* W  = m ? Wo  : Wn;
    float* SQ = m ? sqo : sqn;
    const float* V  = Vmat + m * 4096;
    const float* mu = means + m * 64;
    const float* sv = sval + m * 64;
    __shared__ float Vs[4096];
    __shared__ float xr[4 * 64];
    __shared__ float red[256];
    int tid = threadIdx.x;
    for (int i = tid; i < 4096; i += 256) Vs[i] = V[i];
    int r = tid >> 6, c = tid & 63;
    int row = rb * 4 + r;
    xr[r * 64 + c] = X[(size_t)row * DIMS + c] - mu[c];
    __syncthreads();
    float acc = 0.f;
#pragma unroll
    for (int k = 0; k < 64; ++k) acc += xr[r * 64 + k] * Vs[k * 64 + c];
    float w = acc * rsqrtf(sv[c] + 1e-5f);
    W[(size_t)row * DIMS + c] = w;
    red[tid] = w * w;
    __syncthreads();
    for (int off = 32; off > 0; off >>= 1) {
        if (c < off) red[tid] += red[tid + off];
        __syncthreads();
    }
    if (c == 0) SQ[row] = red[tid];
}

// ----------------------------------- fused Gram + distance + smooth-L1 (WMMA)
// One 128x128 block of the (symmetric) distance matrices per workgroup; only
// the upper block-triangle is launched, off-diagonal blocks weighted x2.
__global__ void loss_kernel(const float* __restrict__ Wn, const float* __restrict__ Wo,
                            const float* __restrict__ sqn, const float* __restrict__ sqo,
                            double* __restrict__ partials) {
    // triangular block id -> (bi, bj), bj >= bi
    int bid = blockIdx.x;
    int bi = 0, rem = bid;
    while (rem >= (NBLK - bi)) { rem -= (NBLK - bi); ++bi; }
    int bj = bi + rem;
    int iBase = bi * TILE, jBase = bj * TILE;

    extern __shared__ float smem[];
    float* sAn  = smem;                 // [128][68]  Wn rows I
    float* sBn  = sAn + TILE * LSTR;    // [128][68]  Wn rows J
    float* sAo  = sBn + TILE * LSTR;
    float* sBo  = sAo + TILE * LSTR;
    float* sqAn = sBo + TILE * LSTR;    // [128]
    float* sqBn = sqAn + TILE;
    float* sqAo = sqBn + TILE;
    float* sqBo = sqAo + TILE;

    for (int t = threadIdx.x; t < TILE * 16; t += 256) {   // float4 granularity
        int row = t >> 4;
        int c4  = (t & 15) << 2;
        *(float4*)&sAn[row * LSTR + c4] = *(const float4*)&Wn[(size_t)(iBase + row) * DIMS + c4];
        *(float4*)&sBn[row * LSTR + c4] = *(const float4*)&Wn[(size_t)(jBase + row) * DIMS + c4];
        *(float4*)&sAo[row * LSTR + c4] = *(const float4*)&Wo[(size_t)(iBase + row) * DIMS + c4];
        *(float4*)&sBo[row * LSTR + c4] = *(const float4*)&Wo[(size_t)(jBase + row) * DIMS + c4];
    }
    if (threadIdx.x < TILE) {
        sqAn[threadIdx.x] = sqn[iBase + threadIdx.x];
        sqBn[threadIdx.x] = sqn[jBase + threadIdx.x];
        sqAo[threadIdx.x] = sqo[iBase + threadIdx.x];
        sqBo[threadIdx.x] = sqo[jBase + threadIdx.x];
    }
    __syncthreads();

    int lane = threadIdx.x & 31;
    int wave = threadIdx.x >> 5;        // 8 waves -> 8 subtile rows
    int half = lane >> 4;               // 0/1
    int lm   = lane & 15;
    float acc = 0.f;

    for (int st = 0; st < 8; ++st) {    // 8 subtile columns per wave
        v8f gn = {}; v8f go = {};
        const float* An = &sAn[(wave * 16 + lm) * LSTR + 2 * half];
        const float* Bn = &sBn[(st   * 16 + lm) * LSTR + 2 * half];
        const float* Ao = &sAo[(wave * 16 + lm) * LSTR + 2 * half];
        const float* Bo = &sBo[(st   * 16 + lm) * LSTR + 2 * half];
#pragma unroll
        for (int k = 0; k < DIMS; k += 4) {
            v2f an = *(const v2f*)(An + k);
            v2f bn = *(const v2f*)(Bn + k);
            gn = __builtin_amdgcn_wmma_f32_16x16x4_f32(false, an, false, bn,
                                                       (short)0, gn, false, false);
            v2f ao = *(const v2f*)(Ao + k);
            v2f bo = *(const v2f*)(Bo + k);
            go = __builtin_amdgcn_wmma_f32_16x16x4_f32(false, ao, false, bo,
                                                       (short)0, go, false, false);
        }
        // C/D layout: VGPR r -> M = r + 8*half ; N = lane & 15
#pragma unroll
        for (int r = 0; r < 8; ++r) {
            int gi = wave * 16 + r + half * 8;
            int gj = st   * 16 + lm;
            float dn = sqAn[gi] + sqBn[gj] - 2.f * gn[r];
            float dd = sqAo[gi] + sqBo[gj] - 2.f * go[r];
            dn = fmaxf(dn, 0.f);
            dd = fmaxf(dd, 0.f);
            float d = fabsf(dn - dd);
            acc += (d < 1.f) ? 0.5f * d * d : d - 0.5f;
        }
    }

    __syncthreads();
    float* red = sAn;                   // reuse LDS
    red[threadIdx.x] = acc;
    __syncthreads();
    for (int off = 128; off > 0; off >>= 1) {
        if (threadIdx.x < off) red[threadIdx.x] += red[threadIdx.x + off];
        __syncthreads();
    }
    if (threadIdx.x == 0) {
        double wgt = (bi == bj) ? 1.0 : 2.0;
        partials[bid] = (double)red[0] * wgt;
    }
}

// ---------------------------------------------------------------- final reduce
__global__ void final_kernel(const double* __restrict__ partials, float* __restrict__ out) {
    __shared__ double red[256];
    double s = 0.0;
    for (int i = threadIdx.x; i < NTRI; i += 256) s += partials[i];
    red[threadIdx.x] = s;
    __syncthreads();
    for (int off = 128; off > 0; off >>= 1) {
        if (threadIdx.x < off) red[threadIdx.x] += red[threadIdx.x + off];
        __syncthreads();
    }
    if (threadIdx.x == 0)
        out[0] = (float)(red[0] / ((double)N_ROWS * (double)N_ROWS) * 10000.0);
}

// ------------------------------------------------------------------- launcher
extern "C" void kernel_launch(void* const* d_in, const int* in_sizes, int n_in,
                              void* d_out, int out_size, void* d_ws, size_t ws_size,
                              hipStream_t stream) {
    const float* Xn = (const float*)d_in[0];   // outputs
    const float* Xo = (const float*)d_in[1];   // initial_outputs
    float* out = (float*)d_out;

    float* ws = (float*)d_ws;
    float* Wn    = ws;                         // 8192*64
    float* Wo    = Wn + N_ROWS * DIMS;         // 8192*64
    float* sqn   = Wo + N_ROWS * DIMS;         // 8192
    float* sqo   = sqn + N_ROWS;               // 8192
    float* means = sqo + N_ROWS;               // 128
    float* cov   = means + 128;                // 2*4096
    float* Vmat  = cov + 2 * 4096;             // 2*4096
    float* sval  = Vmat + 2 * 4096;            // 128
    size_t foff  = (size_t)((sval + 128) - ws);
    foff = (foff + 1) & ~(size_t)1;            // 8B align for doubles
    double* partials = (double*)(ws + foff);   // NTRI doubles

    mean_kernel  <<<128, 256, 0, stream>>>(Xn, Xo, means);
    cov_kernel   <<<2 * 64 * 64, 256, 0, stream>>>(Xn, Xo, means, cov);
    jacobi_kernel<<<2, 64, 0, stream>>>(cov, Vmat, sval);
    whiten_kernel<<<2 * (N_ROWS / 4), 256, 0, stream>>>(Xn, Xo, means, Vmat, sval,
                                                        Wn, Wo, sqn, sqo);
    size_t lds = (size_t)(4 * TILE * LSTR + 4 * TILE) * sizeof(float); // ~141 KB
    loss_kernel  <<<NTRI, 256, lds, stream>>>(Wn, Wo, sqn, sqo, partials);
    final_kernel <<<1, 256, 0, stream>>>(partials, out);
}